// EwaldModelWrapper_28845000359976
// MI455X (gfx1250) — compile-verified
//
#include <hip/hip_runtime.h>
#include <math.h>

typedef __attribute__((ext_vector_type(2))) float v2f;
typedef __attribute__((ext_vector_type(8))) float v8f;

#define NB 16
#define NA 4000
#define NT (NB*NA)
#define NK 64
#define G729 729
#define G2 736          // padded to 46 g-tiles of 16
#define GT 46
#define CH 5            // atom chunks per system for structure-factor partials
#define CHUNK 800       // atoms per chunk (50 WMMA atom-tiles)
#define CTILES 50
#define ATILES 250      // 4000/16 atom tiles per system
#define ALPHA 0.3f
#define KCUT2 1.0f
#define PI_F 3.14159265358979323846f
#define TWO_PI_F 6.283185307179586f
#define INV_SQRT_PI 0.5641895835477563f
#define COULOMB 14.399645351950548f

// workspace offsets (floats)
#define OFF_RECIP  0
#define OFF_U4     256
#define OFF_COEF   (OFF_U4 + NT*4)          // 256256
#define OFF_SPART  (OFF_COEF + NB*G2)       // 268032
#define OFF_WTAB   (OFF_SPART + NB*CH*G2*2) // 385792
#define OFF_EATOM  (OFF_WTAB + NB*G2*2)     // 409344

// ---------- K0: per-system reciprocal matrix + coef table ----------
__global__ void k0_coef(const float* __restrict__ cell, float* __restrict__ ws) {
    int b = blockIdx.x;
    __shared__ float sR[9];
    __shared__ float sInv2V;
    if (threadIdx.x == 0) {
        const float* c = cell + b * 9;
        float a00=c[0],a01=c[1],a02=c[2],a10=c[3],a11=c[4],a12=c[5],a20=c[6],a21=c[7],a22=c[8];
        float det = a00*(a11*a22-a12*a21) - a01*(a10*a22-a12*a20) + a02*(a10*a21-a11*a20);
        float id = 1.0f/det;
        float inv[9];
        inv[0]=(a11*a22-a12*a21)*id; inv[1]=(a02*a21-a01*a22)*id; inv[2]=(a01*a12-a02*a11)*id;
        inv[3]=(a12*a20-a10*a22)*id; inv[4]=(a00*a22-a02*a20)*id; inv[5]=(a02*a10-a00*a12)*id;
        inv[6]=(a10*a21-a11*a20)*id; inv[7]=(a01*a20-a00*a21)*id; inv[8]=(a00*a11-a01*a10)*id;
        // recip[j][d] = 2*pi * inv[d][j]   (rows are reciprocal vectors b_j)
        for (int j = 0; j < 3; ++j)
            for (int d = 0; d < 3; ++d)
                sR[j*3+d] = TWO_PI_F * inv[d*3+j];
        sInv2V = 1.0f / (2.0f * fabsf(det));
        float* rw = ws + OFF_RECIP + b*12;
        for (int i = 0; i < 9; ++i) rw[i] = sR[i];
    }
    __syncthreads();
    float* coefW = ws + OFF_COEF + b*G2;
    for (int g = threadIdx.x; g < G2; g += blockDim.x) {
        float cf = 0.0f;
        if (g < G729) {
            float fx = (float)(g/81 - 4);
            float fy = (float)((g/9)%9 - 4);
            float fz = (float)(g%9 - 4);
            float kx = fx*sR[0] + fy*sR[3] + fz*sR[6];
            float ky = fx*sR[1] + fy*sR[4] + fz*sR[7];
            float kz = fx*sR[2] + fy*sR[5] + fz*sR[8];
            float k2 = kx*kx + ky*ky + kz*kz;
            if (k2 > 1e-10f && k2 <= KCUT2)
                cf = 4.0f*PI_F*__expf(-k2/(4.0f*ALPHA*ALPHA))/k2 * sInv2V;
        }
        coefW[g] = cf;
    }
}

// ---------- K0b: u = pos . recip^T, packed with charge ----------
__global__ void k0b_u(const float* __restrict__ pos, const float* __restrict__ q,
                      float* __restrict__ ws) {
    int t = blockIdx.x*blockDim.x + threadIdx.x;
    if (t >= NT) return;
    int b = t / NA;
    const float* r = ws + OFF_RECIP + b*12;
    float x = pos[t*3], y = pos[t*3+1], z = pos[t*3+2];
    float u0 = x*r[0] + y*r[1] + z*r[2];
    float u1 = x*r[3] + y*r[4] + z*r[5];
    float u2 = x*r[6] + y*r[7] + z*r[8];
    float4* u4 = (float4*)(ws + OFF_U4);
    u4[t] = make_float4(u0, u1, u2, q[t]);
}

// ---------- K1: real-space Ewald, whole system staged in LDS ----------
__global__ void __launch_bounds__(256) k1_real(
        const float* __restrict__ pos, const float* __restrict__ q,
        const float* __restrict__ cell, const int* __restrict__ nm,
        const int* __restrict__ shifts, const int* __restrict__ nn,
        float* __restrict__ ws) {
    __shared__ float4 sys[NA];   // 64000 B: whole system pos+q
    __shared__ float sC[9];
    int b   = blockIdx.x >> 4;   // 16 blocks of 256 per system
    int blk = blockIdx.x & 15;
    int tbase = b * NA;
    for (int i = threadIdx.x; i < NA; i += blockDim.x) {
        int t = tbase + i;
        sys[i] = make_float4(pos[t*3], pos[t*3+1], pos[t*3+2], q[t]);
    }
    if (threadIdx.x < 9) sC[threadIdx.x] = cell[b*9 + threadIdx.x];
    __syncthreads();
    int ia = blk*256 + threadIdx.x;
    if (ia >= NA) return;
    int t = tbase + ia;
    float4 me = sys[ia];
    int nnt = nn[t];
    const int* nmr = nm + (size_t)t * NK;
    const int* shr = shifts + (size_t)t * NK * 3;
    float acc = 0.0f;
    for (int k = 0; k < NK; ++k) {
        int j = nmr[k];
        bool m = (j >= 0) && (k < nnt);
        int jl = m ? (j - tbase) : 0;
        float s0 = (float)shr[k*3+0], s1 = (float)shr[k*3+1], s2 = (float)shr[k*3+2];
        float4 pj = sys[jl];
        float dx = pj.x + s0*sC[0] + s1*sC[3] + s2*sC[6] - me.x;
        float dy = pj.y + s0*sC[1] + s1*sC[4] + s2*sC[7] - me.y;
        float dz = pj.z + s0*sC[2] + s1*sC[5] + s2*sC[8] - me.z;
        float d2 = dx*dx + dy*dy + dz*dz;
        float d  = sqrtf(m ? d2 : 1.0f);
        acc += m ? me.w * pj.w * erfcf(ALPHA * d) / d : 0.0f;
    }
    ws[OFF_EATOM + t] = 0.5f*acc - ALPHA*INV_SQRT_PI*me.w*me.w;
}

// ---------- K2: structure factors via WMMA phase tiles ----------
__global__ void __launch_bounds__(256) k2_sfac(float* __restrict__ ws) {
    __shared__ float4 sU[CHUNK];      // 12.8 KB chunk of (u0,u1,u2,q)
    int b  = blockIdx.x / CH;
    int ch = blockIdx.x % CH;
    const float4* u4 = (const float4*)(ws + OFF_U4);
    int base = b*NA + ch*CHUNK;
    for (int i = threadIdx.x; i < CHUNK; i += blockDim.x) sU[i] = u4[base + i];
    __syncthreads();
    int lane = threadIdx.x & 31;
    int wv   = threadIdx.x >> 5;
    int lo   = lane & 15;
    bool hi  = lane >= 16;
    float* spart = ws + OFF_SPART + (size_t)(b*CH + ch) * G2 * 2;
    for (int gt = wv; gt < GT; gt += 8) {
        int g = gt*16 + lo;
        float fx = 0.f, fy = 0.f, fz = 0.f;
        if (g < G729) {
            fx = (float)(g/81 - 4);
            fy = (float)((g/9)%9 - 4);
            fz = (float)(g%9 - 4);
        }
        v2f bop; bop.x = hi ? fz : fx; bop.y = hi ? 0.0f : fy;   // B 4x16, K rows (nx,ny,nz,0)
        float sre = 0.0f, sim = 0.0f;
        for (int at = 0; at < CTILES; ++at) {
            float4 ua = sU[at*16 + lo];
            v2f aop; aop.x = hi ? ua.z : ua.x; aop.y = hi ? 0.0f : ua.y; // A 16x4 = (u0,u1,u2,0)
            v8f c = {0.f,0.f,0.f,0.f,0.f,0.f,0.f,0.f};
            v8f d = __builtin_amdgcn_wmma_f32_16x16x4_f32(
                        false, aop, false, bop, (short)0, c, false, false);
            int abase = at*16 + (hi ? 8 : 0);
            #pragma unroll
            for (int v = 0; v < 8; ++v) {
                float sn, cn;
                __sincosf(d[v], &sn, &cn);
                float qv = sU[abase + v].w;
                sre += qv * cn;
                sim += qv * sn;
            }
        }
        sre += __shfl_xor(sre, 16, 32);   // combine the two 16-lane halves (same g)
        sim += __shfl_xor(sim, 16, 32);
        if (!hi) { spart[g*2+0] = sre; spart[g*2+1] = sim; }
    }
}

// ---------- K3: fixed-order partial reduce + apply coef ----------
__global__ void k3_w(float* __restrict__ ws) {
    int b = blockIdx.x;
    const float* coefW = ws + OFF_COEF + b*G2;
    float* wtab = ws + OFF_WTAB + (size_t)b * G2 * 2;
    for (int g = threadIdx.x; g < G2; g += blockDim.x) {
        float sre = 0.f, sim = 0.f;
        for (int ch = 0; ch < CH; ++ch) {
            const float* sp = ws + OFF_SPART + (size_t)(b*CH + ch) * G2 * 2;
            sre += sp[g*2+0];
            sim += sp[g*2+1];
        }
        float cf = coefW[g];
        wtab[g*2+0] = cf * sre;
        wtab[g*2+1] = cf * sim;
    }
}

// ---------- K4: per-atom reciprocal energy via WMMA phase tiles ----------
__global__ void __launch_bounds__(256) k4_recip(float* __restrict__ ws) {
    __shared__ float sW[G2*2];        // 5.9 KB coef*S weights for this system
    int b   = blockIdx.x >> 5;        // 32 blocks (of 8 waves) per system
    int blk = blockIdx.x & 31;
    const float* wtab = ws + OFF_WTAB + (size_t)b * G2 * 2;
    for (int i = threadIdx.x; i < G2*2; i += blockDim.x) sW[i] = wtab[i];
    __syncthreads();
    int lane = threadIdx.x & 31;
    int wv   = threadIdx.x >> 5;
    int tile = blk*8 + wv;
    if (tile >= ATILES) return;
    int lo  = lane & 15;
    bool hi = lane >= 16;
    const float4* u4 = (const float4*)(ws + OFF_U4);
    int tbase = b*NA + tile*16;
    float4 ua = u4[tbase + lo];
    v2f aop; aop.x = hi ? ua.z : ua.x; aop.y = hi ? 0.0f : ua.y;
    float acc[8];
    #pragma unroll
    for (int v = 0; v < 8; ++v) acc[v] = 0.0f;
    for (int gt = 0; gt < GT; ++gt) {
        int g = gt*16 + lo;
        float fx = 0.f, fy = 0.f, fz = 0.f;
        if (g < G729) {
            fx = (float)(g/81 - 4);
            fy = (float)((g/9)%9 - 4);
            fz = (float)(g%9 - 4);
        }
        v2f bop; bop.x = hi ? fz : fx; bop.y = hi ? 0.0f : fy;
        v8f c = {0.f,0.f,0.f,0.f,0.f,0.f,0.f,0.f};
        v8f d = __builtin_amdgcn_wmma_f32_16x16x4_f32(
                    false, aop, false, bop, (short)0, c, false, false);
        float wre = sW[g*2+0], wim = sW[g*2+1];
        #pragma unroll
        for (int v = 0; v < 8; ++v) {
            float sn, cn;
            __sincosf(d[v], &sn, &cn);
            acc[v] += cn*wre + sn*wim;
        }
    }
    #pragma unroll
    for (int v = 0; v < 8; ++v) {          // sum over the 16 g-lanes of each half
        float s = acc[v];
        s += __shfl_xor(s, 1, 32);
        s += __shfl_xor(s, 2, 32);
        s += __shfl_xor(s, 4, 32);
        s += __shfl_xor(s, 8, 32);
        acc[v] = s;
    }
    if (lo < 8) {
        float out = acc[0];
        #pragma unroll
        for (int v = 1; v < 8; ++v) out = (lo == v) ? acc[v] : out;
        int t = tbase + lo + (hi ? 8 : 0);
        float qv = u4[t].w;
        ws[OFF_EATOM + t] += qv * out;     // unique writer per atom -> deterministic
    }
}

// ---------- K5: fixed-order per-system reduction ----------
__global__ void __launch_bounds__(256) k5_out(const float* __restrict__ ws,
                                              float* __restrict__ out) {
    __shared__ float red[256];
    int b = blockIdx.x;
    const float* ea = ws + OFF_EATOM + (size_t)b * NA;
    float s = 0.0f;
    for (int i = threadIdx.x; i < NA; i += 256) s += ea[i];
    red[threadIdx.x] = s;
    __syncthreads();
    for (int st = 128; st > 0; st >>= 1) {
        if (threadIdx.x < st) red[threadIdx.x] += red[threadIdx.x + st];
        __syncthreads();
    }
    if (threadIdx.x == 0) out[b] = red[0] * COULOMB;
}

extern "C" void kernel_launch(void* const* d_in, const int* in_sizes, int n_in,
                              void* d_out, int out_size, void* d_ws, size_t ws_size,
                              hipStream_t stream) {
    const float* positions = (const float*)d_in[0];
    const float* charges   = (const float*)d_in[1];
    const float* cell      = (const float*)d_in[2];
    const int*   nm        = (const int*)d_in[3];
    const int*   shifts    = (const int*)d_in[4];
    const int*   nn        = (const int*)d_in[5];
    float* ws  = (float*)d_ws;
    float* out = (float*)d_out;

    k0_coef <<<NB, 256, 0, stream>>>(cell, ws);
    k0b_u   <<<(NT + 255)/256, 256, 0, stream>>>(positions, charges, ws);
    k1_real <<<NB*16, 256, 0, stream>>>(positions, charges, cell, nm, shifts, nn, ws);
    k2_sfac <<<NB*CH, 256, 0, stream>>>(ws);
    k3_w    <<<NB, 256, 0, stream>>>(ws);
    k4_recip<<<NB*32, 256, 0, stream>>>(ws);
    k5_out  <<<NB, 256, 0, stream>>>(ws, out);
}